// DenseGridSampler_11269994184717
// MI455X (gfx1250) — compile-verified
//
#include <hip/hip_runtime.h>
#include <hip/hip_bf16.h>
#include <stdint.h>

// ---------------- problem constants (jit-static in the reference) ----------
#define N_RAYS      32768
#define BLOCK       128          // 4 wave32s per block; 256 blocks
#define NGRID       128
#define NCASC       5
#define NSAMPLE     128
#define MARCH_STEPS 512
#define CHUNK       16           // independent occupancy gathers in flight
#define NEAR_DIST   0.2f
#define CONE        (1.0f/256.0f)
#define MIN_STEPF   0.02706329338f          // sqrt(3)*16/1024
#define MAX_STEPF   (16.0f*MIN_STEPF)
#define HALF_EXT    8.0f                    // 0.5 * 2^(NCASC-1)

typedef __attribute__((ext_vector_type(4))) unsigned int v4u;
typedef __attribute__((ext_vector_type(8))) int          v8i;
typedef __attribute__((ext_vector_type(4))) int          v4i;

#if defined(__has_builtin)
#  if __has_builtin(__builtin_amdgcn_tensor_load_to_lds)
#    define HAVE_TDM 1
#  endif
#endif

#if defined(HAVE_TDM)
// 1-D TDM copy: n_dwords DWORDs from global `gsrc` into LDS byte-offset lds_off.
// D# group0: count=1, lds_addr, 57-bit global addr, type=2 ("image").
// D# group1: data_size=4B, tensor_dim0=n_dwords, tensor_dim1=1, tile_dim0=n_dwords,
//            tensor_dim0_stride=n_dwords, no multicast / padding / iterate.
__device__ __forceinline__ void tdm_copy_1d(const void* gsrc, uint32_t lds_off,
                                            uint32_t n_dwords) {
  uint64_t ga = (uint64_t)(uintptr_t)gsrc;
  v4u g0;
  g0.x = 1u;                                         // count=1, user mode
  g0.y = lds_off;                                    // LDS byte address
  g0.z = (uint32_t)ga;                               // global addr [31:0]
  g0.w = (uint32_t)((ga >> 32) & 0x01FFFFFFu)        // global addr [56:32]
         | (2u << 30);                               // type=2
  v8i g1;
  g1[0] = (int)(2u << 16);                           // data_size=2 (4 bytes)
  g1[1] = (int)((n_dwords & 0xFFFFu) << 16);         // tensor_dim0[15:0]
  g1[2] = (int)(((n_dwords >> 16) & 0xFFFFu)         // tensor_dim0[31:16]
         | (1u << 16));                              // tensor_dim1 = 1
  g1[3] = (int)((n_dwords & 0xFFFFu) << 16);         // tile_dim0 = n_dwords
  g1[4] = 0;                                         // tile_dim1/2 unused
  g1[5] = (int)n_dwords;                             // tensor_dim0_stride lo
  g1[6] = 0;
  g1[7] = 0;
  v4i z4 = {0, 0, 0, 0};
#if __clang_major__ >= 23
  v8i z8 = {0, 0, 0, 0, 0, 0, 0, 0};
  __builtin_amdgcn_tensor_load_to_lds(g0, g1, z4, z4, z8, 0);
#else
  __builtin_amdgcn_tensor_load_to_lds(g0, g1, z4, z4, 0);
#endif
}
#endif

__device__ __forceinline__ int cell_coord(float p, float inv_ext) {
  float u  = p * inv_ext + 0.5f;       // extent is a power of two -> exact recip
  int   i  = (int)(u * 128.0f);        // v_cvt_i32_f32: trunc-toward-zero, clamps
  i = i < 0 ? 0 : i;
  return i > (NGRID - 1) ? (NGRID - 1) : i;
}

// exact ceil(log2(x)) for x >= 1.0f, no transcendentals:
// biased_exp - 127, +1 if mantissa bits are nonzero.
__device__ __forceinline__ int ceil_log2_ge1(float x) {
  uint32_t u = __float_as_uint(x);
  int e = (int)(u >> 23) - 127;
  return e + ((u & 0x007FFFFFu) ? 1 : 0);
}

__global__ __launch_bounds__(BLOCK) void dense_grid_march(
    const float* __restrict__ rays_o, const float* __restrict__ rays_d,
    const unsigned char* __restrict__ bitfield, float* __restrict__ out) {
  __shared__ float s_o[BLOCK * 3];
  __shared__ float s_d[BLOCK * 3];

  const int tid = threadIdx.x;
  const int ray = blockIdx.x * BLOCK + tid;

#if defined(HAVE_TDM)
  if (tid == 0) {
    const float* go = rays_o + (size_t)blockIdx.x * BLOCK * 3;
    const float* gd = rays_d + (size_t)blockIdx.x * BLOCK * 3;
    tdm_copy_1d(go, (uint32_t)(uintptr_t)(void*)s_o, BLOCK * 3);
    tdm_copy_1d(gd, (uint32_t)(uintptr_t)(void*)s_d, BLOCK * 3);
    __builtin_amdgcn_s_wait_tensorcnt(0);
  }
  __syncthreads();
  const float ox = s_o[tid * 3 + 0], oy = s_o[tid * 3 + 1], oz = s_o[tid * 3 + 2];
  const float rx = s_d[tid * 3 + 0], ry = s_d[tid * 3 + 1], rz = s_d[tid * 3 + 2];
#else
  const float ox = rays_o[ray * 3 + 0], oy = rays_o[ray * 3 + 1], oz = rays_o[ray * 3 + 2];
  const float rx = rays_d[ray * 3 + 0], ry = rays_d[ray * 3 + 1], rz = rays_d[ray * 3 + 2];
#endif

  // normalize direction
  const float inv_len = 1.0f / sqrtf(rx * rx + ry * ry + rz * rz);
  const float dx = rx * inv_len, dy = ry * inv_len, dz = rz * inv_len;

  // ray / coarsest-AABB intersection
  const float sx = (fabsf(dx) < 1e-9f) ? 1e-9f : dx;
  const float sy = (fabsf(dy) < 1e-9f) ? 1e-9f : dy;
  const float sz = (fabsf(dz) < 1e-9f) ? 1e-9f : dz;
  const float ivx = 1.0f / sx, ivy = 1.0f / sy, ivz = 1.0f / sz;
  const float t1x = (-HALF_EXT - ox) * ivx, t2x = (HALF_EXT - ox) * ivx;
  const float t1y = (-HALF_EXT - oy) * ivy, t2y = (HALF_EXT - oy) * ivy;
  const float t1z = (-HALF_EXT - oz) * ivz, t2z = (HALF_EXT - oz) * ivz;
  float tn = fmaxf(fmaxf(fminf(t1x, t2x), fminf(t1y, t2y)), fminf(t1z, t2z));
  const float tf = fminf(fminf(fmaxf(t1x, t2x), fmaxf(t1y, t2y)), fmaxf(t1z, t2z));
  tn = fmaxf(tn, NEAR_DIST);
  const bool hit = tf > tn;

  // output regions (tuple concatenated flat: pts, dirs, dt, numsteps)
  float* pts  = out;
  float* dirs = out + (size_t)N_RAYS * NSAMPLE * 3;
  float* dts  = dirs + (size_t)N_RAYS * NSAMPLE * 3;
  int*   ns   = (int*)(dts + (size_t)N_RAYS * NSAMPLE);

  int   count = 0;
  float t     = tn;
  if (hit) {
    for (int c = 0; c < MARCH_STEPS / CHUNK; ++c) {
      if (t >= tf || count >= NSAMPLE) break;   // exact: t is monotone increasing
      float tv[CHUNK], dtv[CHUNK];
      int   cell[CHUNK];
      // address chain is independent of load results -> batch CHUNK gathers
#pragma unroll
      for (int i = 0; i < CHUNK; ++i) {
        const float dt = fminf(fmaxf(t * CONE, MIN_STEPF), MAX_STEPF);
        tv[i] = t; dtv[i] = dt;
        const float px = ox + t * dx, py = oy + t * dy, pz = oz + t * dz;
        const float ma = fmaxf(fmaxf(fabsf(px), fabsf(py)), fabsf(pz));
        int mip = ceil_log2_ge1(fmaxf(2.0f * ma, 1.0f));
        mip = mip > (NCASC - 1) ? (NCASC - 1) : mip;   // >=0 by construction
        const float inv_ext = __int_as_float((127 - mip) << 23);   // exact 2^-mip
        const int ix = cell_coord(px, inv_ext);
        const int iy = cell_coord(py, inv_ext);
        const int iz = cell_coord(pz, inv_ext);
        cell[i] = ((mip * NGRID + ix) * NGRID + iy) * NGRID + iz;
        t += dt;
      }
      unsigned char occ[CHUNK];
#pragma unroll
      for (int i = 0; i < CHUNK; ++i) occ[i] = bitfield[cell[i]];  // 16 loads in flight
#pragma unroll
      for (int i = 0; i < CHUNK; ++i) {
        if (occ[i] && (tv[i] < tf) && (count < NSAMPLE)) {
          const int s = ray * NSAMPLE + count;
          const float tt = tv[i];
          // streaming (NT) stores: outputs are write-once, keep L2 for bitfield
          __builtin_nontemporal_store(ox + tt * dx, &pts[3 * s + 0]);
          __builtin_nontemporal_store(oy + tt * dy, &pts[3 * s + 1]);
          __builtin_nontemporal_store(oz + tt * dz, &pts[3 * s + 2]);
          __builtin_nontemporal_store(dx, &dirs[3 * s + 0]);
          __builtin_nontemporal_store(dy, &dirs[3 * s + 1]);
          __builtin_nontemporal_store(dz, &dirs[3 * s + 2]);
          __builtin_nontemporal_store(dtv[i], &dts[s]);
          ++count;
        }
      }
    }
  }
  // zero-fill tail so every output element is written each launch (harness
  // poisons d_out once and replays without re-poisoning)
  for (int j = count; j < NSAMPLE; ++j) {
    const int s = ray * NSAMPLE + j;
    __builtin_nontemporal_store(0.0f, &pts[3 * s + 0]);
    __builtin_nontemporal_store(0.0f, &pts[3 * s + 1]);
    __builtin_nontemporal_store(0.0f, &pts[3 * s + 2]);
    __builtin_nontemporal_store(0.0f, &dirs[3 * s + 0]);
    __builtin_nontemporal_store(0.0f, &dirs[3 * s + 1]);
    __builtin_nontemporal_store(0.0f, &dirs[3 * s + 2]);
    __builtin_nontemporal_store(0.0f, &dts[s]);
  }
  __builtin_nontemporal_store(count, &ns[2 * ray + 0]);
  __builtin_nontemporal_store(ray * NSAMPLE, &ns[2 * ray + 1]);
}

extern "C" void kernel_launch(void* const* d_in, const int* in_sizes, int n_in,
                              void* d_out, int out_size, void* d_ws, size_t ws_size,
                              hipStream_t stream) {
  (void)in_sizes; (void)n_in; (void)out_size; (void)d_ws; (void)ws_size;
  const float*         rays_o   = (const float*)d_in[0];
  const float*         rays_d   = (const float*)d_in[1];
  const unsigned char* bitfield = (const unsigned char*)d_in[2];
  float*               out      = (float*)d_out;
  dim3 grid(N_RAYS / BLOCK), block(BLOCK);
  hipLaunchKernelGGL(dense_grid_march, grid, block, 0, stream,
                     rays_o, rays_d, bitfield, out);
}